// CustomLoss_6330781795106
// MI455X (gfx1250) — compile-verified
//
#include <hip/hip_runtime.h>
#include <hip/hip_bf16.h>
#include <math.h>

// CustomLoss for MI455X / gfx1250.
// Memory-bound: 128 MiB single pass over y_pred (~5.7us at 23.3 TB/s).
// CDNA5 path: global_load_async_to_lds_b128 (ASYNCcnt) stages each 16KB row
// into LDS; all softmax statistics computed in one pass from registers/LDS.

#define CLS   4096
#define BATCH 8192
#define TPB   256   // 8 wave32s per workgroup

__global__ __launch_bounds__(TPB) void row_stats_kernel(
    const float* __restrict__ y_pred,
    const int*   __restrict__ labels,   // harness passes integer inputs as int32
    float* __restrict__ ws)             // [0..B)=nl_i  [B..2B)=ce2_i  [2B..3B)=log1p(pyD^2)
{
    __shared__ __align__(16) float smem[CLS];   // 16 KB row stage
    __shared__ float red_f[8];
    __shared__ int   red_i[8];

    const int tid = threadIdx.x;
    const int row = blockIdx.x;

    // ---- Async copy: global row -> LDS (4 x b128 per thread = 16 KB/block).
    // GV mode: 64-bit vaddr pair, inst offset applies to both LDS and global.
    unsigned           lds = (unsigned)(size_t)smem + (unsigned)tid * 16u;
    unsigned long long ga  = (unsigned long long)(size_t)(y_pred + (size_t)row * CLS)
                           + (unsigned long long)tid * 16ull;
    asm volatile("global_load_async_to_lds_b128 %0, %1, off"              :: "v"(lds), "v"(ga) : "memory");
    asm volatile("global_load_async_to_lds_b128 %0, %1, off offset:4096"  :: "v"(lds), "v"(ga) : "memory");
    asm volatile("global_load_async_to_lds_b128 %0, %1, off offset:8192"  :: "v"(lds), "v"(ga) : "memory");
    asm volatile("global_load_async_to_lds_b128 %0, %1, off offset:12288" :: "v"(lds), "v"(ga) : "memory");
    asm volatile("s_wait_asynccnt 0x0" ::: "memory");
    __syncthreads();

    // ---- Each thread pulls its 16 elements into VGPRs (ds_load_b128 x4).
    float4 v[4];
#pragma unroll
    for (int k = 0; k < 4; ++k)
        v[k] = *reinterpret_cast<const float4*>(&smem[k * 1024 + tid * 4]);

    // ---- Phase A: row max + argmax (first occurrence, like jnp.argmax).
    float m = -INFINITY; int ai = CLS;
#pragma unroll
    for (int k = 0; k < 4; ++k) {
        const float* e = reinterpret_cast<const float*>(&v[k]);
#pragma unroll
        for (int j = 0; j < 4; ++j) {
            float x = e[j];
            int col = k * 1024 + tid * 4 + j;
            if (x > m) { m = x; ai = col; }   // in-thread cols ascend -> first occurrence
        }
    }
    for (int off = 16; off > 0; off >>= 1) {
        float om = __shfl_xor(m, off, 32);
        int   oi = __shfl_xor(ai, off, 32);
        if (om > m || (om == m && oi < ai)) { m = om; ai = oi; }
    }
    const int wv = tid >> 5, lane = tid & 31;
    if (lane == 0) { red_f[wv] = m; red_i[wv] = ai; }
    __syncthreads();
    float M = red_f[0]; int AI = red_i[0];
#pragma unroll
    for (int i = 1; i < 8; ++i) {
        float om = red_f[i]; int oi = red_i[i];
        if (om > M || (om == M && oi < AI)) { M = om; AI = oi; }
    }
    __syncthreads();

    // ---- Phase B: S = sum(exp(x - M)).
    float s = 0.0f;
#pragma unroll
    for (int k = 0; k < 4; ++k) {
        const float* e = reinterpret_cast<const float*>(&v[k]);
#pragma unroll
        for (int j = 0; j < 4; ++j) s += expf(e[j] - M);
    }
    for (int off = 16; off > 0; off >>= 1) s += __shfl_xor(s, off, 32);
    if (lane == 0) red_f[wv] = s;
    __syncthreads();
    float S = 0.0f;
#pragma unroll
    for (int i = 0; i < 8; ++i) S += red_f[i];
    __syncthreads();

    // ---- Phase C: cnt = #{j : p_j >= 1/C}  <=>  #{j : x >= M + ln(S) - ln(C)}.
    // (lower clip 1e-7 < 1/4096 and upper clip 1.0 cannot change the count)
    const float thresh = M + logf(S) - logf((float)CLS);
    int c = 0;
#pragma unroll
    for (int k = 0; k < 4; ++k) {
        const float* e = reinterpret_cast<const float*>(&v[k]);
#pragma unroll
        for (int j = 0; j < 4; ++j) c += (e[j] >= thresh) ? 1 : 0;
    }
    for (int off = 16; off > 0; off >>= 1) c += __shfl_xor(c, off, 32);
    if (lane == 0) red_i[wv] = c;
    __syncthreads();
    int CNT = 0;
#pragma unroll
    for (int i = 0; i < 8; ++i) CNT += red_i[i];

    // ---- Per-row epilogue (y[label] is already staged in LDS).
    if (tid == 0) {
        int   lab  = labels[row];
        float ylab = smem[lab];
        float py = expf(ylab - M) / S;
        py = fminf(fmaxf(py, 1e-7f), 1.0f);
        float ce   = logf(fminf(fmaxf(1.0f - py, 1e-7f), 1.0f));
        float nl_i = (py - 1.0f) * ce;
        float pmax = fminf(fmaxf(1.0f / S, 1e-7f), 1.0f);   // max p = exp(0)/S, clipped
        float pyD  = (CNT < 2) ? pmax : 0.0f;
        float w_i  = log1pf(pyD * pyD);                     // log(1 + pyD^2)
        float ce2_i = (float)AI * logf(py);                 // pred_label * log(p[label])
        ws[row]             = nl_i;
        ws[BATCH + row]     = ce2_i;
        ws[2 * BATCH + row] = w_i;
    }
}

__global__ __launch_bounds__(TPB) void finalize_kernel(
    const float* __restrict__ ws, float* __restrict__ out)
{
    __shared__ float s_nl[TPB], s_c2[TPB], s_w[TPB];
    const int t = threadIdx.x;
    float a = 0.0f, b = 0.0f, c = 0.0f;
    for (int i = t; i < BATCH; i += TPB) {          // fixed order -> deterministic
        a += ws[i];
        b += ws[BATCH + i];
        c += ws[2 * BATCH + i];
    }
    s_nl[t] = a; s_c2[t] = b; s_w[t] = c;
    __syncthreads();
    for (int off = TPB / 2; off > 0; off >>= 1) {
        if (t < off) {
            s_nl[t] += s_nl[t + off];
            s_c2[t] += s_c2[t + off];
            s_w[t]  += s_w[t + off];
        }
        __syncthreads();
    }
    if (t == 0) {
        float weight = expf(s_w[0]);                // prod(1 + pyD^2)
        float pl = -weight * s_c2[0];
        out[0] = s_nl[0] / (float)BATCH + 0.01f * pl;
    }
}

extern "C" void kernel_launch(void* const* d_in, const int* in_sizes, int n_in,
                              void* d_out, int out_size, void* d_ws, size_t ws_size,
                              hipStream_t stream) {
    (void)in_sizes; (void)n_in; (void)out_size; (void)ws_size;
    const float* y_pred = (const float*)d_in[0];
    const int*   labels = (const int*)d_in[1];
    float*       out    = (float*)d_out;
    float*       wsf    = (float*)d_ws;   // needs 3*8192*4 = 96 KB

    row_stats_kernel<<<BATCH, TPB, 0, stream>>>(y_pred, labels, wsf);
    finalize_kernel<<<1, TPB, 0, stream>>>(wsf, out);
}